// CollectConcat_13048110645918
// MI455X (gfx1250) — compile-verified
//
#include <hip/hip_runtime.h>
#include <stdint.h>

#define B_   4
#define C_   720
#define H_   128
#define W_   128
#define P_   9
#define OFF_ 80
#define HW_  (H_ * W_)

typedef __attribute__((ext_vector_type(4))) unsigned int u32x4;
typedef __attribute__((ext_vector_type(8))) int          i32x8;
typedef __attribute__((ext_vector_type(4))) int          i32x4;
// float2 with 4-byte alignment so the compiler legally emits global_load_b64
// at any dword-aligned address (DWORD alignment mode allows 4B-aligned b64).
typedef float f2 __attribute__((ext_vector_type(2), aligned(4)));

__device__ __forceinline__ int iclamp(int v, int lo, int hi) {
    return v < lo ? lo : (v > hi ? hi : v);
}

__global__ __launch_bounds__(256) void bilinear_sample_kernel(
    const float* __restrict__ x,        // (B, C, H, W)   -> keep L2-resident
    const float* __restrict__ loc,      // (B, 2P, H, W)  -> read-once, NT
    const float* __restrict__ bias,     // (C,)           -> staged via TDM
    float*       __restrict__ out)      // (B, C, H, W)   -> streaming, NT
{
    __shared__ float sbias[OFF_];

    const int tid = blockIdx.x * 256 + threadIdx.x;
    const int hw  = tid & (HW_ - 1);    // HW_ = 2^14
    const int bp  = tid >> 14;
    const int p   = bp % P_;            // uniform per block (16384 % 256 == 0)
    const int b   = bp / P_;

    // ---- Stage bias[p*OFF_ .. +79] into LDS with the Tensor Data Mover ----
    if (threadIdx.x < 32) {             // wave 0 issues the TDM descriptor
        const uint64_t ga  = (uint64_t)(uintptr_t)(bias + p * OFF_);
        const uint32_t lds = (uint32_t)(uintptr_t)(&sbias[0]);

        u32x4 g0;
        g0[0] = 1u;                                     // count = 1 (valid D#)
        g0[1] = lds;                                    // lds_addr
        g0[2] = (uint32_t)(ga & 0xFFFFFFFFu);           // global_addr[31:0]
        g0[3] = (uint32_t)((ga >> 32) & 0x01FFFFFFu)    // global_addr[56:32]
              | (2u << 30);                             // type = 2 (image)

        // D# group1 (256b): data_size=2(4B), tensor_dim0=80, tensor_dim1=1,
        // tile_dim0=80, tile_dim1=1, tile_dim2=0, stride0=80, stride1=0
        const uint64_t q0 = ((uint64_t)2u << 16)                    // data_size
                          | ((uint64_t)(OFF_ & 0xFFFF) << 48);      // dim0 lo16
        const uint64_t q1 = (uint64_t)((OFF_ >> 16) & 0xFFFF)       // dim0 hi16
                          | ((uint64_t)1u << 16)                    // tensor_dim1
                          | ((uint64_t)OFF_ << 48);                 // tile_dim0
        const uint64_t q2 = (uint64_t)1u                            // tile_dim1
                          | ((uint64_t)(OFF_ & 0xFFFFFFFFu) << 32); // stride0 lo32
        const uint64_t q3 = 0;                                      // stride0 hi / stride1

        i32x8 g1;
        g1[0] = (int)(uint32_t)(q0);       g1[1] = (int)(uint32_t)(q0 >> 32);
        g1[2] = (int)(uint32_t)(q1);       g1[3] = (int)(uint32_t)(q1 >> 32);
        g1[4] = (int)(uint32_t)(q2);       g1[5] = (int)(uint32_t)(q2 >> 32);
        g1[6] = (int)(uint32_t)(q3);       g1[7] = (int)(uint32_t)(q3 >> 32);

        i32x4 gz4 = {0, 0, 0, 0};          // groups 2/3 unused (<=2D tensor)
        i32x8 gz8 = {0, 0, 0, 0, 0, 0, 0, 0};

        // amdgpu-toolchain (clang-23) 6-arg form:
        // (u32x4, i32x8, i32x4, i32x4, i32x8, i32 cpol)
        __builtin_amdgcn_tensor_load_to_lds(g0, g1, gz4, gz4, gz8, 0);
        __builtin_amdgcn_s_wait_tensorcnt(0);
    }
    __syncthreads();

    // ---- Per-location bilinear setup (amortized over 80 channels) ----
    const float yq = __builtin_nontemporal_load(
        loc + ((size_t)(b * (2 * P_) + 2 * p)     ) * HW_ + hw);
    const float xq = __builtin_nontemporal_load(
        loc + ((size_t)(b * (2 * P_) + 2 * p + 1)) * HW_ + hw);

    const float y0f = floorf(yq), x0f = floorf(xq);
    const float dy = yq - y0f,    dx = xq - x0f;
    const int   y0 = (int)y0f,    x0 = (int)x0f;

    const bool vy0 = (y0 >= 0)  && (y0 < H_);
    const bool vy1 = (y0 >= -1) && (y0 < H_ - 1);
    const bool vx0 = (x0 >= 0)  && (x0 < W_);
    const bool vx1 = (x0 >= -1) && (x0 < W_ - 1);

    const float w00 = (1.f - dy) * (1.f - dx) * ((vy0 && vx0) ? 1.f : 0.f);
    const float w01 = (1.f - dy) * dx         * ((vy0 && vx1) ? 1.f : 0.f);
    const float w10 = dy * (1.f - dx)         * ((vy1 && vx0) ? 1.f : 0.f);
    const float w11 = dy * dx                 * ((vy1 && vx1) ? 1.f : 0.f);

    const int yr0 = iclamp(y0,     0, H_ - 1);
    const int yr1 = iclamp(y0 + 1, 0, H_ - 1);
    const int xb  = iclamp(x0,     0, W_ - 2);   // pair-load base
    const bool leftx = (x0 == xb);               // is pair[0] the x0 sample?

    // Swap the WEIGHTS once (identical for all 80 channels) instead of
    // selecting values per channel: invalid corners carry weight 0, so any
    // in-bounds garbage lane contributes 0.
    const float wa0 = leftx ? w00 : w01;   // applies to row0 pair.x
    const float wb0 = leftx ? w01 : w00;   // applies to row0 pair.y
    const float wa1 = leftx ? w10 : w11;   // applies to row1 pair.x
    const float wb1 = leftx ? w11 : w10;   // applies to row1 pair.y

    const int off0 = yr0 * W_ + xb;
    const int off1 = yr1 * W_ + xb;

    const float* plane = x   + ((size_t)(b * C_ + p * OFF_)) * HW_;
    float*       optr  = out + ((size_t)(b * C_ + p * OFF_)) * HW_ + hw;

    // Warm the first channel's lines while the wave finishes setup.
    __builtin_prefetch(plane + off0, 0, 0);
    __builtin_prefetch(plane + off1, 0, 0);

    // ---- Channel loop: 2 paired b64 gathers + 4 FMA + 1 NT store ----
    // Channel offsets (c*HW_ + off) fit the 24-bit signed immediate, so the
    // unrolled body issues loads with constant offsets from one base.
    #pragma unroll 8
    for (int c = 0; c < OFF_; ++c) {
        const float* pc = plane + (size_t)c * HW_;

        const f2 r0 = *reinterpret_cast<const f2*>(pc + off0);
        const f2 r1 = *reinterpret_cast<const f2*>(pc + off1);

        float r = sbias[c];
        r = fmaf(wb1, r1.y, r);
        r = fmaf(wa1, r1.x, r);
        r = fmaf(wb0, r0.y, r);
        r = fmaf(wa0, r0.x, r);

        __builtin_nontemporal_store(r, optr + (size_t)c * HW_);
    }
}

extern "C" void kernel_launch(void* const* d_in, const int* in_sizes, int n_in,
                              void* d_out, int out_size, void* d_ws, size_t ws_size,
                              hipStream_t stream) {
    const float* x    = (const float*)d_in[0];   // (B, C, H, W)
    const float* loc  = (const float*)d_in[1];   // (B, 2P, H, W)
    const float* bias = (const float*)d_in[2];   // (C,)
    float* out        = (float*)d_out;

    const int n_loc  = B_ * P_ * HW_;            // 589,824 threads, 1 per location
    const int blocks = n_loc / 256;              // 2304
    bilinear_sample_kernel<<<blocks, 256, 0, stream>>>(x, loc, bias, out);
}